// VoxelBackBone8x_13237089206443
// MI455X (gfx1250) — compile-verified
//
#include <hip/hip_runtime.h>
#include <math.h>
#include <stdint.h>

// CDNA5 / gfx1250: wave32, WMMA (no MFMA), async global->LDS copies.
typedef float v2f __attribute__((ext_vector_type(2)));
typedef float v4f __attribute__((ext_vector_type(4)));
typedef float v8f __attribute__((ext_vector_type(8)));
typedef int   v4i __attribute__((ext_vector_type(4)));

#define WAVE_SZ 32

#if __has_builtin(__builtin_amdgcn_global_load_async_to_lds_b128)
#define ASYNC_LDS_COPY 1
typedef __attribute__((address_space(1))) v4i* gv4i_ptr;
typedef __attribute__((address_space(3))) v4i* lv4i_ptr;
#else
#define ASYNC_LDS_COPY 0
#endif

__device__ __forceinline__ void wait_async_all() {
#if ASYNC_LDS_COPY
#if __has_builtin(__builtin_amdgcn_s_wait_asynccnt)
    __builtin_amdgcn_s_wait_asynccnt(0);
#else
    asm volatile("s_wait_asynccnt 0x0" ::: "memory");
#endif
#endif
}

// Copy one CIN x COUT weight tile (row-major in global) into LDS with a
// +4-float row pad (bank de-conflict for the two K-halves of the wave).
// Uses GLOBAL_LOAD_ASYNC_TO_LDS_B128 (ASYNCcnt) when available.
template <int CIN, int COUT>
__device__ __forceinline__ void stage_weights(const float* __restrict__ g,
                                              float* __restrict__ l) {
    constexpr int STRIDE = COUT + 4;
    const int wave = threadIdx.x >> 5;
    const int lane = threadIdx.x & (WAVE_SZ - 1);
    const int nw = blockDim.x >> 5;
    for (int r = wave; r < CIN; r += nw) {
        for (int e = lane * 4; e < COUT; e += WAVE_SZ * 4) {
            const float* src = g + r * COUT + e;
            float* dst = l + r * STRIDE + e;
#if ASYNC_LDS_COPY
            // Generic->AS1 for global is value-preserving; generic->AS3 LDS
            // address is the low 32 bits of the flat address (ISA 10.2).
            __builtin_amdgcn_global_load_async_to_lds_b128(
                (gv4i_ptr)(uintptr_t)src,
                (lv4i_ptr)(uintptr_t)(uint32_t)(uintptr_t)dst,
                0, 0);
#else
            *(v4f*)dst = *(const v4f*)src;
#endif
        }
    }
}

// ---------------------------------------------------------------------------
// Dense coordinate -> point-index maps
// ---------------------------------------------------------------------------
__global__ void fill_i32(int* __restrict__ p, int n, int val) {
    int i = blockIdx.x * blockDim.x + threadIdx.x;
    if (i < n) p[i] = val;
}

__global__ void build_map(const int* __restrict__ coords, int n,
                          int gd, int gh, int gw, int* __restrict__ map) {
    int i = blockIdx.x * blockDim.x + threadIdx.x;
    if (i >= n) return;
    int b = coords[i * 4 + 0];
    int z = coords[i * 4 + 1];
    int y = coords[i * 4 + 2];
    int x = coords[i * 4 + 3];
    map[((b * gd + z) * gh + y) * gw + x] = i;
}

// ---------------------------------------------------------------------------
// Implicit-gather sparse conv via V_WMMA_F32_16X16X4_F32.
// One wave: 16 output points x COUT channels. Per-offset weight tiles are
// double-buffered in LDS (async prefetch of offset o+1 overlaps WMMA of o).
//   A (16x4 f32): lanes 0-15 -> M=lane, K={kb,kb+1}; lanes 16-31 -> K={kb+2,kb+3}
//   B (4x16 f32): lane&15 -> N, lane>>4 selects the K pair (mirrors A)
//   C/D (16x16 f32, 8 VGPRs): vgpr j -> row j + 8*(lane>>4), col = lane&15
// All waves run the full offset loop (no early return) so __syncthreads() and
// WMMA (EXEC must be all 1s) stay in uniform control flow.
// ---------------------------------------------------------------------------
template <int CIN, int COUT>
__global__ void __launch_bounds__(256)
sp_conv_wmma(const float* __restrict__ fin, float* __restrict__ fout,
             const int* __restrict__ ocoords, int Nout,
             const int* __restrict__ map, int gd, int gh, int gw,
             const float* __restrict__ w,
             int k0, int k1, int k2,
             int s0, int s1, int s2,
             int p0, int p1, int p2) {
    constexpr int NT = COUT / 16;      // N tiles of 16 channels
    constexpr int KS = CIN / 4;        // K steps of 4 channels
    constexpr int STRIDE = COUT + 4;   // padded LDS row stride (floats)
    constexpr int TILE = CIN * COUT;   // global floats per offset tile

    __shared__ __align__(16) float lds_w[2][CIN * STRIDE];

    const int lane = threadIdx.x & (WAVE_SZ - 1);
    const int wave = threadIdx.x >> 5;
    const int mt = blockIdx.x * (blockDim.x >> 5) + wave;  // 16-point tile
    const int ntiles_m = (Nout + 15) >> 4;
    const bool tileValid = mt < ntiles_m;

    const int m = lane & 15;      // output-point row for A gather
    const int khalf = lane >> 4;  // 0 -> K {0,1}, 1 -> K {2,3}
    const int n = lane & 15;      // output channel column for B / C

    const int pidx = mt * 16 + m;
    const bool mvalid = tileValid && (pidx < Nout);
    int cb = 0, cz = 0, cy = 0, cx = 0;
    if (mvalid) {
        cb = ocoords[pidx * 4 + 0];
        cz = ocoords[pidx * 4 + 1];
        cy = ocoords[pidx * 4 + 2];
        cx = ocoords[pidx * 4 + 3];
    }

    v8f acc[NT] = {};

    const int nOff = k0 * k1 * k2;

    // Prime buffer 0 with offset 0's weight tile.
    stage_weights<CIN, COUT>(w, lds_w[0]);
    wait_async_all();
    __syncthreads();

    int o0 = 0, o1 = 0, o2 = 0;  // incremental offset decode
    for (int o = 0; o < nOff; ++o) {
        const int buf = o & 1;
        // Prefetch next offset tile into the other buffer (overlaps compute).
        if (o + 1 < nOff)
            stage_weights<CIN, COUT>(w + (size_t)(o + 1) * TILE, lds_w[buf ^ 1]);

        const int iz = cz * s0 + o0 - p0;
        const int iy = cy * s1 + o1 - p1;
        const int ix = cx * s2 + o2 - p2;
        int gidx = -1;
        if (mvalid && iz >= 0 && iz < gd && iy >= 0 && iy < gh && ix >= 0 &&
            ix < gw)
            gidx = map[((cb * gd + iz) * gh + iy) * gw + ix];
        const float* __restrict__ fi =
            (gidx >= 0) ? (fin + (size_t)gidx * CIN) : nullptr;
        const float* __restrict__ lw = lds_w[buf];

#pragma unroll
        for (int ks = 0; ks < KS; ++ks) {
            const int kb = ks * 4 + 2 * khalf;
            v2f a;
            if (fi)
                a = *(const v2f*)(fi + kb);  // 8B-aligned: kb even, CIN even
            else
                a = (v2f){0.0f, 0.0f};
#pragma unroll
            for (int nt = 0; nt < NT; ++nt) {
                v2f bf;
                bf.x = lw[(kb + 0) * STRIDE + nt * 16 + n];
                bf.y = lw[(kb + 1) * STRIDE + nt * 16 + n];
                acc[nt] = __builtin_amdgcn_wmma_f32_16x16x4_f32(
                    false, a, false, bf, (short)0, acc[nt], false, false);
            }
        }

        // Ensure prefetch landed and all waves finished reading lds_w[buf]
        // before it gets overwritten next iteration.
        wait_async_all();
        __syncthreads();

        if (++o2 == k2) {
            o2 = 0;
            if (++o1 == k1) {
                o1 = 0;
                ++o0;
            }
        }
    }

#pragma unroll
    for (int nt = 0; nt < NT; ++nt) {
#pragma unroll
        for (int j = 0; j < 8; ++j) {
            const int row = mt * 16 + j + 8 * khalf;
            if (tileValid && row < Nout)
                fout[(size_t)row * COUT + nt * 16 + n] = acc[nt][j];
        }
    }
}

// ---------------------------------------------------------------------------
// BatchNorm statistics: one block per channel, deterministic LDS tree reduce.
// stats[c] = mean, stats[C + c] = var (E[x^2] - mu^2)
// ---------------------------------------------------------------------------
__global__ void __launch_bounds__(256)
bn_stats(const float* __restrict__ x, int N, int C, float* __restrict__ stats) {
    __shared__ float s_sum[256];
    __shared__ float s_sq[256];
    const int c = blockIdx.x;
    float sum = 0.0f, sq = 0.0f;
    for (int i = threadIdx.x; i < N; i += blockDim.x) {
        float v = x[(size_t)i * C + c];
        sum += v;
        sq += v * v;
    }
    s_sum[threadIdx.x] = sum;
    s_sq[threadIdx.x] = sq;
    __syncthreads();
    for (int off = 128; off > 0; off >>= 1) {
        if ((int)threadIdx.x < off) {
            s_sum[threadIdx.x] += s_sum[threadIdx.x + off];
            s_sq[threadIdx.x] += s_sq[threadIdx.x + off];
        }
        __syncthreads();
    }
    if (threadIdx.x == 0) {
        float mu = s_sum[0] / (float)N;
        stats[c] = mu;
        stats[C + c] = s_sq[0] / (float)N - mu * mu;
    }
}

__global__ void __launch_bounds__(256)
bn_apply(float* __restrict__ x, int N, int C,
         const float* __restrict__ stats,
         const float* __restrict__ gamma, const float* __restrict__ beta) {
    int i = blockIdx.x * blockDim.x + threadIdx.x;
    if (i >= N * C) return;
    const int c = i % C;
    const float mu = stats[c];
    const float var = stats[C + c];
    const float scale = gamma[c] * rsqrtf(var + 1e-3f);
    float v = (x[i] - mu) * scale + beta[c];
    x[i] = v > 0.0f ? v : 0.0f;
}

// ---------------------------------------------------------------------------
// Host orchestration
// ---------------------------------------------------------------------------
extern "C" void kernel_launch(void* const* d_in, const int* in_sizes, int n_in,
                              void* d_out, int out_size, void* d_ws, size_t ws_size,
                              hipStream_t stream) {
    (void)n_in; (void)out_size; (void)ws_size;

    const float* vf = (const float*)d_in[0];
    const float* W[12];
    for (int i = 0; i < 12; ++i) W[i] = (const float*)d_in[1 + i];
    const float* gamma = (const float*)d_in[13];
    const float* beta  = (const float*)d_in[14];
    const int* c1 = (const int*)d_in[15];
    const int* c2 = (const int*)d_in[16];
    const int* c3 = (const int*)d_in[17];
    const int* c4 = (const int*)d_in[18];
    const int* c5 = (const int*)d_in[19];
    const int N1 = in_sizes[15] / 4;
    const int N2 = in_sizes[16] / 4;
    const int N3 = in_sizes[17] / 4;
    const int N4 = in_sizes[18] / 4;
    const int N5 = in_sizes[19] / 4;

    // Scratch carving (256B aligned)
    char* base = (char*)d_ws;
    size_t off = 0;
    auto carve = [&](size_t nbytes) -> void* {
        void* p = base + off;
        off += (nbytes + 255) & ~(size_t)255;
        return p;
    };
    const int G1 = 2 * 41 * 160 * 160;
    const int G2 = 2 * 21 * 80 * 80;
    const int G3 = 2 * 11 * 40 * 40;
    const int G4 = 2 * 5 * 20 * 20;
    int* map1 = (int*)carve(sizeof(int) * (size_t)G1);
    int* map2 = (int*)carve(sizeof(int) * (size_t)G2);
    int* map3 = (int*)carve(sizeof(int) * (size_t)G3);
    int* map4 = (int*)carve(sizeof(int) * (size_t)G4);
    float* f1a = (float*)carve(sizeof(float) * (size_t)N1 * 16);
    float* f1b = (float*)carve(sizeof(float) * (size_t)N1 * 16);
    float* f2a = (float*)carve(sizeof(float) * (size_t)N2 * 32);
    float* f2b = (float*)carve(sizeof(float) * (size_t)N2 * 32);
    float* f3a = (float*)carve(sizeof(float) * (size_t)N3 * 64);
    float* f3b = (float*)carve(sizeof(float) * (size_t)N3 * 64);
    float* f4a = (float*)carve(sizeof(float) * (size_t)N4 * 64);
    float* f4b = (float*)carve(sizeof(float) * (size_t)N4 * 64);
    float* stats = (float*)carve(sizeof(float) * 256);

    // Build coordinate->index maps (rebuilt identically every call: deterministic)
    fill_i32<<<(G1 + 255) / 256, 256, 0, stream>>>(map1, G1, -1);
    fill_i32<<<(G2 + 255) / 256, 256, 0, stream>>>(map2, G2, -1);
    fill_i32<<<(G3 + 255) / 256, 256, 0, stream>>>(map3, G3, -1);
    fill_i32<<<(G4 + 255) / 256, 256, 0, stream>>>(map4, G4, -1);
    build_map<<<(N1 + 255) / 256, 256, 0, stream>>>(c1, N1, 41, 160, 160, map1);
    build_map<<<(N2 + 255) / 256, 256, 0, stream>>>(c2, N2, 21, 80, 80, map2);
    build_map<<<(N3 + 255) / 256, 256, 0, stream>>>(c3, N3, 11, 40, 40, map3);
    build_map<<<(N4 + 255) / 256, 256, 0, stream>>>(c4, N4, 5, 20, 20, map4);

#define RUN_LAYER(CIN, COUT, FIN, FOUT, OC, NOUT, MAP, GD, GH, GW, WT,      \
                  K0, K1, K2, S0, S1_, S2_, P0, P1, P2, GOFF)               \
    do {                                                                    \
        int tiles_ = ((NOUT) + 15) / 16;                                    \
        int blocks_ = (tiles_ + 7) / 8; /* 8 waves (256 thr) per block */   \
        sp_conv_wmma<CIN, COUT><<<blocks_, 256, 0, stream>>>(               \
            FIN, FOUT, OC, NOUT, MAP, GD, GH, GW, WT,                       \
            K0, K1, K2, S0, S1_, S2_, P0, P1, P2);                          \
        bn_stats<<<COUT, 256, 0, stream>>>(FOUT, NOUT, COUT, stats);        \
        bn_apply<<<((NOUT) * (COUT) + 255) / 256, 256, 0, stream>>>(        \
            FOUT, NOUT, COUT, stats, gamma + (GOFF), beta + (GOFF));        \
    } while (0)

    RUN_LAYER(4, 16,  vf,  f1a, c1, N1, map1, 41, 160, 160, W[0], 3, 3, 3, 1, 1, 1, 1, 1, 1, 0);
    RUN_LAYER(16, 16, f1a, f1b, c1, N1, map1, 41, 160, 160, W[1], 3, 3, 3, 1, 1, 1, 1, 1, 1, 16);
    RUN_LAYER(16, 32, f1b, f2a, c2, N2, map1, 41, 160, 160, W[2], 3, 3, 3, 2, 2, 2, 1, 1, 1, 32);
    RUN_LAYER(32, 32, f2a, f2b, c2, N2, map2, 21, 80, 80,  W[3], 3, 3, 3, 1, 1, 1, 1, 1, 1, 64);
    RUN_LAYER(32, 32, f2b, f2a, c2, N2, map2, 21, 80, 80,  W[4], 3, 3, 3, 1, 1, 1, 1, 1, 1, 96);
    RUN_LAYER(32, 64, f2a, f3a, c3, N3, map2, 21, 80, 80,  W[5], 3, 3, 3, 2, 2, 2, 1, 1, 1, 128);
    RUN_LAYER(64, 64, f3a, f3b, c3, N3, map3, 11, 40, 40,  W[6], 3, 3, 3, 1, 1, 1, 1, 1, 1, 192);
    RUN_LAYER(64, 64, f3b, f3a, c3, N3, map3, 11, 40, 40,  W[7], 3, 3, 3, 1, 1, 1, 1, 1, 1, 256);
    RUN_LAYER(64, 64, f3a, f4a, c4, N4, map3, 11, 40, 40,  W[8], 3, 3, 3, 2, 2, 2, 0, 1, 1, 320);
    RUN_LAYER(64, 64, f4a, f4b, c4, N4, map4, 5, 20, 20,   W[9], 3, 3, 3, 1, 1, 1, 1, 1, 1, 384);
    RUN_LAYER(64, 64, f4b, f4a, c4, N4, map4, 5, 20, 20,   W[10], 3, 3, 3, 1, 1, 1, 1, 1, 1, 448);
    RUN_LAYER(64, 128, f4a, (float*)d_out, c5, N5, map4, 5, 20, 20, W[11], 3, 1, 1, 2, 1, 1, 0, 0, 0, 512);

#undef RUN_LAYER
}